// SGC_7103875907621
// MI455X (gfx1250) — compile-verified
//
#include <hip/hip_runtime.h>

typedef float v2f __attribute__((ext_vector_type(2)));
typedef float v8f __attribute__((ext_vector_type(8)));
typedef unsigned int v4u __attribute__((ext_vector_type(4)));
typedef int v4i __attribute__((ext_vector_type(4)));
typedef int v8i __attribute__((ext_vector_type(8)));

// ---------------- zero ----------------
__global__ void SGC_zero_f32(float* __restrict__ p, int n) {
    int i = blockIdx.x * blockDim.x + threadIdx.x;
    if (i < n) p[i] = 0.0f;
}

// ---------------- copy x [N x F] -> padded [Npad x KPAD] (pad stays zero) --------
__global__ void SGC_copy_pad(const float* __restrict__ src, float* __restrict__ dst,
                             int N, int F, int KPAD) {
    int i = blockIdx.x * blockDim.x + threadIdx.x;
    if (i >= N * F) return;
    int node = i / F, f = i - node * F;
    dst[(size_t)node * KPAD + f] = src[i];
}

// ---------------- W1 [F x M] -> transposed padded [M x KPAD] ----------------
__global__ void SGC_transpose_pad(const float* __restrict__ W1, float* __restrict__ W1pT,
                                  int F, int M, int KPAD) {
    int i = blockIdx.x * blockDim.x + threadIdx.x;
    if (i >= F * M) return;
    int k = i / M, c = i - k * M;
    W1pT[(size_t)c * KPAD + k] = W1[i];
}

// ---------------- degree: deg[col]++ for edges + self loops ----------------
__global__ void SGC_degree(const int* __restrict__ col, float* __restrict__ deg,
                           int E, int N) {
    int i = blockIdx.x * blockDim.x + threadIdx.x;
    if (i >= E + N) return;
    int c = (i < E) ? col[i] : (i - E);
    atomicAdd(&deg[c], 1.0f);
}

// ---------------- dinv = rsqrt(deg) (deg >= 1 due to self loops) ----------------
__global__ void SGC_dinv(const float* __restrict__ deg, float* __restrict__ dinv, int N) {
    int i = blockIdx.x * blockDim.x + threadIdx.x;
    if (i < N) dinv[i] = rsqrtf(fmaxf(deg[i], 1e-12f));
}

// ------- vectorized hop: one thread per (edge, float4 chunk), KPAD = 4*CH -------
// coalesced b128 gather of x[row], 4x f32 atomic scatter into x[col]; pad column
// carries zeros so no masking is needed anywhere.
template <int CH>
__global__ void SGC_prop_vec(const float* __restrict__ xin, float* __restrict__ xout,
                             const int* __restrict__ row, const int* __restrict__ col,
                             const float* __restrict__ dinv, int E, int N) {
    const int KPAD = 4 * CH;
    int i = blockIdx.x * blockDim.x + threadIdx.x;
    if (i >= (E + N) * CH) return;
    int e = i / CH;                      // CH is a literal -> mul/shift, no v_div
    int q = i - e * CH;
    int r = (e < E) ? row[e] : (e - E);
    int c = (e < E) ? col[e] : (e - E);
    float nrm = dinv[r] * dinv[c];
    const float4 s4 = *(const float4*)(xin + (size_t)r * KPAD + q * 4);
    float* __restrict__ dst = xout + (size_t)c * KPAD + q * 4;
    atomicAdd(dst + 0, nrm * s4.x);
    atomicAdd(dst + 1, nrm * s4.y);
    atomicAdd(dst + 2, nrm * s4.z);
    atomicAdd(dst + 3, nrm * s4.w);
}

// ---------------- scalar hop (generic shapes) ----------------
__global__ void SGC_prop(const float* __restrict__ xin, int inStride,
                         float* __restrict__ xout, int outStride,
                         const int* __restrict__ row, const int* __restrict__ col,
                         const float* __restrict__ dinv, int E, int N, int F) {
    int i = blockIdx.x * blockDim.x + threadIdx.x;
    if (i >= E + N) return;
    int r = (i < E) ? row[i] : (i - E);
    int c = (i < E) ? col[i] : (i - E);
    float nrm = dinv[r] * dinv[c];
    const float* __restrict__ src = xin + (size_t)r * inStride;
    float* __restrict__ dst = xout + (size_t)c * outStride;
    for (int f = 0; f < F; ++f) atomicAdd(&dst[f], nrm * src[f]);
}

// ---------------- fused GEMM (x @ W1 + b1) + global_add_pool -------------------
// Block = 8 waves; block owns node tile [mBase,mBase+16) x all MM cols, wave w owns
// cols [16w,16w+16). The shared 16xKPAD A-tile is DMA'd to LDS once by wave 0 via
// the Tensor Data Mover; each wave then runs a branch-free chain of
// V_WMMA_F32_16X16X4_F32 (A from LDS via ds_load_2addr_b64, B via b64 loads from
// the transposed W1) and pools the tile into pooled[batch[node]][col].
template <int KPAD, int MM>
__global__ __launch_bounds__(256)
void SGC_gemm_pool_tdm(const float* __restrict__ x,       // [Npad x KPAD], zero padded
                       const float* __restrict__ W1pT,    // [MM x KPAD], zero padded
                       const float* __restrict__ b1, const int* __restrict__ batch,
                       float* __restrict__ pooled, int N, int Npad) {
    __shared__ float ldsA[16 * KPAD];

    const int tid   = threadIdx.x;
    const int wave  = tid >> 5;             // 0..7 -> col tile
    const int lane  = tid & 31;
    const int half  = lane >> 4;            // 0 or 1
    const int lm    = lane & 15;
    const int mBase = blockIdx.x * 16;      // node tile

    if (wave == 0) {
        // ---- Tensor DMA descriptor: 2D tile [16 rows x KPAD cols] of f32 ----
        unsigned ldsOff = (unsigned)(size_t)(&ldsA[0]);   // flat shared addr[31:0] = LDS offset
        unsigned long long ga =
            (unsigned long long)(size_t)x + (unsigned long long)mBase * (KPAD * 4ull);
        v4u g0;
        g0.x = 1u;                                        // count=1, user descriptor
        g0.y = ldsOff;                                    // lds_addr
        g0.z = (unsigned)(ga & 0xFFFFFFFFu);              // global_addr[31:0]
        g0.w = (unsigned)((ga >> 32) & 0x1FFFFFFu) | 0x80000000u; // global_addr[56:32] | type=2
        v8i g1;
        g1[0] = 0x00020000;                               // wg_mask=0, data_size=2 (4B), no flags
        g1[1] = (KPAD & 0xFFFF) << 16;                    // tensor_dim0[15:0]
        g1[2] = ((KPAD >> 16) & 0xFFFF) | ((Npad & 0xFFFF) << 16);   // dim0 hi | tensor_dim1 lo
        g1[3] = ((Npad >> 16) & 0xFFFF) | ((KPAD & 0xFFFF) << 16);   // dim1 hi | tile_dim0=KPAD
        g1[4] = 16;                                       // tile_dim1=16, tile_dim2=0
        g1[5] = KPAD;                                     // tensor_dim0_stride[31:0]
        g1[6] = 0;
        g1[7] = 0;
        v4i gz4 = {0, 0, 0, 0};
        v8i gz8 = {0, 0, 0, 0, 0, 0, 0, 0};
        __builtin_amdgcn_tensor_load_to_lds(g0, g1, gz4, gz4, gz8, 0);
        __builtin_amdgcn_s_wait_tensorcnt(0);
    }
    __syncthreads();

    const int colIdx = wave * 16 + lm;
    const float* __restrict__ bp = W1pT + (size_t)colIdx * KPAD;
    const float* __restrict__ ap = &ldsA[lm * KPAD];

    v8f c = {0.f, 0.f, 0.f, 0.f, 0.f, 0.f, 0.f, 0.f};
#pragma unroll
    for (int k0 = 0; k0 < KPAD; k0 += 4) {
        const int ka = k0 + 2 * half;
        v2f a = *(const v2f*)(ap + ka);                   // 8B LDS load
        v2f b = *(const v2f*)(bp + ka);                   // 8B global load (transposed W1)
        c = __builtin_amdgcn_wmma_f32_16x16x4_f32(false, a, false, b,
                                                  (short)0, c, false, false);
    }

    const float bias = b1[colIdx];
#pragma unroll
    for (int r = 0; r < 8; ++r) {
        const int node = mBase + r + 8 * half;
        if (node < N) {
            const int g = batch[node];
            atomicAdd(&pooled[(size_t)g * MM + colIdx], c[r] + bias);
        }
    }
}

// ---------------- generic fallback (non-reference shapes): scalar dot + pool ----
__global__ void SGC_gemm_pool_generic(const float* __restrict__ x, int xStride,
                                      const float* __restrict__ W1,
                                      const float* __restrict__ b1,
                                      const int* __restrict__ batch,
                                      float* __restrict__ pooled, int N, int K, int M) {
    int i = blockIdx.x * blockDim.x + threadIdx.x;
    if (i >= N * M) return;
    int node = i / M, col = i % M;
    const float* __restrict__ xr = x + (size_t)node * xStride;
    float s = b1[col];
    for (int k = 0; k < K; ++k) s += xr[k] * W1[(size_t)k * M + col];
    atomicAdd(&pooled[(size_t)batch[node] * M + col], s);
}

// ---------------- out[g] = pooled[g,:] . W2 + b2  (one wave per graph) ----------
__global__ void SGC_final(const float* __restrict__ pooled, const float* __restrict__ W2,
                          const float* __restrict__ b2, float* __restrict__ out,
                          int G, int M) {
    int w    = (blockIdx.x * blockDim.x + threadIdx.x) >> 5;
    int lane = threadIdx.x & 31;
    if (w >= G) return;
    const float* __restrict__ p = pooled + (size_t)w * M;
    float s = 0.0f;
    for (int f = lane; f < M; f += 32) s += p[f] * W2[f];
#pragma unroll
    for (int off = 16; off > 0; off >>= 1) s += __shfl_xor(s, off, 32);
    if (lane == 0) out[w] = s + b2[0];
}

extern "C" void kernel_launch(void* const* d_in, const int* in_sizes, int n_in,
                              void* d_out, int out_size, void* d_ws, size_t ws_size,
                              hipStream_t stream) {
    const float* x     = (const float*)d_in[0];
    const int*   ei    = (const int*)  d_in[1];   // [2,E] flattened: rows then cols
    const int*   batch = (const int*)  d_in[2];
    const float* W1    = (const float*)d_in[3];
    const float* b1    = (const float*)d_in[4];
    const float* W2    = (const float*)d_in[5];
    const float* b2    = (const float*)d_in[6];
    float* out = (float*)d_out;

    const int N = in_sizes[2];                // 50000 nodes
    const int E = in_sizes[1] / 2;            // 800000 edges
    const int F = in_sizes[0] / N;            // 75 in-features
    const int M = in_sizes[4];                // 128 hidden features
    const int G = out_size;                   // 512 graphs

    const int KPAD = (F + 4) & ~3;            // 75 -> 76 (>=F+1, multiple of 4)
    const int CH   = KPAD / 4;                // 19 float4 chunks per row
    const int Npad = (N + 15) & ~15;

    // workspace (floats): deg | dinv | bufA | bufB | W1pT | pooled
    // bufA: padded x copy (hop1 input), then re-zeroed, then hop2 output / GEMM input
    // bufB: hop1 output / hop2 input
    float* ws     = (float*)d_ws;
    float* deg    = ws;                                   // N
    float* dinv   = deg  + N;                             // N
    float* bufA   = dinv + N;                             // Npad*KPAD
    float* bufB   = bufA + (size_t)Npad * KPAD;           // Npad*KPAD
    float* W1pT   = bufB + (size_t)Npad * KPAD;           // M*KPAD
    float* pooled = W1pT + (size_t)M * KPAD;              // G*M
    const size_t totalZero = (size_t)2 * N + (size_t)2 * Npad * KPAD
                           + (size_t)M * KPAD + (size_t)G * M;

    const int T = 256;
    const int padCnt = Npad * KPAD;

    SGC_zero_f32<<<dim3((unsigned)((totalZero + T - 1) / T)), dim3(T), 0, stream>>>(
        ws, (int)totalZero);

    SGC_copy_pad<<<dim3((N * F + T - 1) / T), dim3(T), 0, stream>>>(x, bufA, N, F, KPAD);
    SGC_transpose_pad<<<dim3((F * M + T - 1) / T), dim3(T), 0, stream>>>(W1, W1pT, F, M, KPAD);

    const int* row = ei;
    const int* col = ei + E;

    SGC_degree<<<dim3((E + N + T - 1) / T), dim3(T), 0, stream>>>(col, deg, E, N);
    SGC_dinv  <<<dim3((N + T - 1) / T),     dim3(T), 0, stream>>>(deg, dinv, N);

    if (CH == 19) {
        // K_HOPS = 2, vectorized: hop1 bufA->bufB, re-zero bufA, hop2 bufB->bufA
        const int PT = (E + N) * 19;
        SGC_prop_vec<19><<<dim3((PT + T - 1) / T), dim3(T), 0, stream>>>(
            bufA, bufB, row, col, dinv, E, N);
        SGC_zero_f32<<<dim3((padCnt + T - 1) / T), dim3(T), 0, stream>>>(bufA, padCnt);
        SGC_prop_vec<19><<<dim3((PT + T - 1) / T), dim3(T), 0, stream>>>(
            bufB, bufA, row, col, dinv, E, N);
    } else {
        SGC_prop<<<dim3((E + N + T - 1) / T), dim3(T), 0, stream>>>(
            bufA, KPAD, bufB, KPAD, row, col, dinv, E, N, F);
        SGC_zero_f32<<<dim3((padCnt + T - 1) / T), dim3(T), 0, stream>>>(bufA, padCnt);
        SGC_prop<<<dim3((E + N + T - 1) / T), dim3(T), 0, stream>>>(
            bufB, KPAD, bufA, KPAD, row, col, dinv, E, N, F);
    }

    if (F == 75 && M == 128) {
        // fast path: TDM-staged A tile, 8 waves x 19 branch-free WMMAs per block
        SGC_gemm_pool_tdm<76, 128><<<dim3(Npad / 16), dim3(256), 0, stream>>>(
            bufA, W1pT, b1, batch, pooled, N, Npad);
    } else {
        SGC_gemm_pool_generic<<<dim3(((size_t)N * M + T - 1) / T), dim3(T), 0, stream>>>(
            bufA, KPAD, W1, b1, batch, pooled, N, F, M);
    }

    SGC_final<<<dim3((G * 32 + T - 1) / T), dim3(T), 0, stream>>>(pooled, W2, b2, out, G, M);
}